// KernelDensityEstimator_66133906423951
// MI455X (gfx1250) — compile-verified
//
#include <hip/hip_runtime.h>

typedef __attribute__((ext_vector_type(2))) float v2f;
typedef __attribute__((ext_vector_type(8))) float v8f;

#define B_ROWS 2048
#define N_KER  16384
#define DIM    16
#define NSPLIT 32
#define SLICE_N (N_KER / NSPLIT)          // 512 columns per slice
#define TILES_PER_SLICE (SLICE_N / 16)    // 32 tiles
#define LOG2PI_F 1.8378770664093453f
#define LOG2E_F  1.4426950408889634f      // log2(e)
#define LN2_F    0.6931471805599453f

// Raw v_exp_f32: no range fixup, no pre-multiply. Args here are always <= 0;
// below ~-126 the true term is < 2^-126 and flush-to-zero is exactly right.
__device__ __forceinline__ float fast_exp2(float a) {
  return __builtin_amdgcn_exp2f(a);
}

// ---------------- kernel 1: per-kernel constants (base-2 domain) + per-row |x|^2 ----
__global__ void kde_prep(const float* __restrict__ x,
                         const float* __restrict__ mu,
                         const float* __restrict__ st,
                         const float* __restrict__ w,
                         float* __restrict__ p2,
                         float* __restrict__ q2,
                         float* __restrict__ xx) {
  int i = blockIdx.x * blockDim.x + threadIdx.x;
  if (i < N_KER) {
    const float4* r = (const float4*)(mu + i * DIM);
    float mm = 0.f;
#pragma unroll
    for (int j = 0; j < 4; ++j) {
      float4 v = r[j];
      mm += v.x * v.x + v.y * v.y + v.z * v.z + v.w * v.w;
    }
    float ls = st[i];                         // log(sigma) == sigmatilde
    float pp = -0.5f * __expf(-2.f * ls);     // -0.5 / sigma^2
    float cn = -0.5f * (2.f * DIM * ls + DIM * LOG2PI_F) + __logf(w[i]);
    // exponent (nat log) = pp*(|x|^2 - 2*dot) + (pp*|mu|^2 + cn)
    // store scaled by log2(e) so the hot loop works in base-2 directly.
    p2[i] = pp * LOG2E_F;
    q2[i] = fmaf(pp, mm, cn) * LOG2E_F;
  }
  if (i < B_ROWS) {
    const float4* r = (const float4*)(x + i * DIM);
    float s2 = 0.f;
#pragma unroll
    for (int j = 0; j < 4; ++j) {
      float4 v = r[j];
      s2 += v.x * v.x + v.y * v.y + v.z * v.z + v.w * v.w;
    }
    xx[i] = s2;
  }
}

// ---------------- kernel 2: fused WMMA distance + online base-2 logsumexp ----------
// One wave (32 threads) per (16-row x-tile, N-slice).
__global__ __launch_bounds__(32) void kde_main(const float* __restrict__ x,
                                               const float* __restrict__ mu,
                                               const float* __restrict__ p2,
                                               const float* __restrict__ q2,
                                               const float* __restrict__ xx,
                                               float2* __restrict__ partial) {
  const int lane = threadIdx.x;
  const int lo = lane & 15;
  const int hi = lane >> 4;
  const int mbase = blockIdx.x * 16;
  const int slice = blockIdx.y;
  const int nbase0 = slice * SLICE_N;

  // A tile (16 rows of x, K=16 as 4 chunks of K=4).
  // ISA layout: lane holds K pair (k0 + 2*hi, k0 + 2*hi + 1) of row lo.
  const float* xrow = x + (size_t)(mbase + lo) * DIM + 2 * hi;
  v2f a0 = *(const v2f*)(xrow + 0);
  v2f a1 = *(const v2f*)(xrow + 4);
  v2f a2 = *(const v2f*)(xrow + 8);
  v2f a3 = *(const v2f*)(xrow + 12);

  // |x|^2 for the 8 rows this lane's C registers cover (M = v + 8*hi).
  float xv[8];
#pragma unroll
  for (int v = 0; v < 8; ++v) xv[v] = xx[mbase + 8 * hi + v];

  float m[8], s[8];
#pragma unroll
  for (int v = 0; v < 8; ++v) { m[v] = -__builtin_inff(); s[v] = 0.f; }

  for (int t = 0; t < TILES_PER_SLICE; ++t) {
    const int nbase = nbase0 + t * 16;
    // B tile: B[k][n] = mu[n][k]; lane holds pair (k0+2*hi, +1) of column lo.
    const float* murow = mu + (size_t)(nbase + lo) * DIM + 2 * hi;
    v2f b0 = *(const v2f*)(murow + 0);
    v2f b1 = *(const v2f*)(murow + 4);
    v2f b2 = *(const v2f*)(murow + 8);
    v2f b3 = *(const v2f*)(murow + 12);

    v8f c = {};
    c = __builtin_amdgcn_wmma_f32_16x16x4_f32(false, a0, false, b0, (short)0, c, false, false);
    c = __builtin_amdgcn_wmma_f32_16x16x4_f32(false, a1, false, b1, (short)0, c, false, false);
    c = __builtin_amdgcn_wmma_f32_16x16x4_f32(false, a2, false, b2, (short)0, c, false, false);
    c = __builtin_amdgcn_wmma_f32_16x16x4_f32(false, a3, false, b3, (short)0, c, false, false);

    const int n = nbase + lo;     // this lane's column within the tile
    const float pp = p2[n];
    const float qq = q2[n];

#pragma unroll
    for (int v = 0; v < 8; ++v) {
      // base-2 exponent: e = log2(e) * [p*(|x|^2 - 2*dot) + q]
      float e = fmaf(pp, fmaf(-2.f, c[v], xv[v]), qq);
      float nm = fmaxf(m[v], e);
      s[v] = fmaf(s[v], fast_exp2(m[v] - nm), fast_exp2(e - nm));
      m[v] = nm;
    }
  }

  // Merge the 16 lanes of each half (each lane held a column-residue class).
#pragma unroll
  for (int v = 0; v < 8; ++v) {
#pragma unroll
    for (int mask = 8; mask >= 1; mask >>= 1) {
      float om = __shfl_xor(m[v], mask, 32);
      float os = __shfl_xor(s[v], mask, 32);
      float nm = fmaxf(m[v], om);
      s[v] = fmaf(s[v], fast_exp2(m[v] - nm), os * fast_exp2(om - nm));
      m[v] = nm;
    }
  }

  if (lo == 0) {
#pragma unroll
    for (int v = 0; v < 8; ++v) {
      int row = mbase + 8 * hi + v;
      partial[(size_t)row * NSPLIT + slice] = make_float2(m[v], s[v]);
    }
  }
}

// ---------------- kernel 3: combine slice partials, back to natural log ----------
__global__ void kde_combine(const float2* __restrict__ partial,
                            float* __restrict__ out) {
  int b = blockIdx.x * blockDim.x + threadIdx.x;
  if (b >= B_ROWS) return;
  float m = -__builtin_inff(), s = 0.f;
#pragma unroll
  for (int j = 0; j < NSPLIT; ++j) {
    float2 ps = partial[(size_t)b * NSPLIT + j];
    float nm = fmaxf(m, ps.x);
    s = fmaf(s, fast_exp2(m - nm), ps.y * fast_exp2(ps.x - nm));
    m = nm;
  }
  out[b] = (m + __log2f(s)) * LN2_F;   // convert base-2 LSE to natural log
}

extern "C" void kernel_launch(void* const* d_in, const int* in_sizes, int n_in,
                              void* d_out, int out_size, void* d_ws, size_t ws_size,
                              hipStream_t stream) {
  const float* x  = (const float*)d_in[0];
  const float* mu = (const float*)d_in[1];
  const float* st = (const float*)d_in[2];
  const float* w  = (const float*)d_in[3];
  float* out = (float*)d_out;

  float* p2 = (float*)d_ws;                 // [N]
  float* q2 = p2 + N_KER;                   // [N]
  float* xx = q2 + N_KER;                   // [B]
  float2* partial = (float2*)(xx + B_ROWS); // [B * NSPLIT], 8-byte aligned

  kde_prep<<<(N_KER + 255) / 256, 256, 0, stream>>>(x, mu, st, w, p2, q2, xx);

  dim3 grid(B_ROWS / 16, NSPLIT);
  kde_main<<<grid, 32, 0, stream>>>(x, mu, p2, q2, xx, partial);

  kde_combine<<<(B_ROWS + 255) / 256, 256, 0, stream>>>(partial, out);
}